// TorchModel_77223511982255
// MI455X (gfx1250) — compile-verified
//
#include <hip/hip_runtime.h>
#include <hip/hip_bf16.h>

// sum(fft2(x)) over all bins == H*W * sum_b x[b,0,0]  (exact DFT identity).
// x: (32, 2048, 2048) f32 contiguous. Result: complex scalar -> {real, 0}.
//
// Single wave32 kernel: each lane loads x[lane*H*W]; a V_WMMA_F32_16X16X4_F32
// with a ones B-matrix performs the 32-way add (A[m,0]=x[m] from lanes 0-15,
// A[m,2]=x[16+m] from lanes 16-31 per the CDNA5 A-matrix 16x4 layout), leaving
// partial sums replicated across D columns; one shfl_xor(16) completes it.

typedef __attribute__((ext_vector_type(2))) float v2f;
typedef __attribute__((ext_vector_type(8))) float v8f;

__global__ __launch_bounds__(32)
void fft2_sum_kernel(const float* __restrict__ x, float* __restrict__ out,
                     int out_size, long long batch_stride) {
    const int lane = threadIdx.x;  // 0..31, one wave32, EXEC all ones

    // Gather the 32 DC-corner samples x[b, 0, 0].
    float v = x[(long long)lane * batch_stride];

    // A-matrix (16x4 f32): VGPR0 = {K=0 | K=2} across lane halves, VGPR1 = 0.
    v2f a;
    a.x = v;
    a.y = 0.0f;

    // B-matrix (4x16 f32) = all ones: D[m,n] = sum_k A[m,k].
    v2f b;
    b.x = 1.0f;
    b.y = 1.0f;

    v8f c = {};
    // D = A x B + 0   (emits v_wmma_f32_16x16x4_f32)
    v8f d = __builtin_amdgcn_wmma_f32_16x16x4_f32(
        /*neg_a=*/false, a, /*neg_b=*/false, b,
        /*c_mod=*/(short)0, c, /*reuse_a=*/false, /*reuse_b=*/false);

    // Lanes 0-15: sum over M=0..7 of column n; lanes 16-31: M=8..15.
    float s = d[0] + d[1] + d[2] + d[3] + d[4] + d[5] + d[6] + d[7];
    s += __shfl_xor(s, 16, 32);  // total = sum_b x[b,0,0] on every lane

    const float scale = 2048.0f * 2048.0f;  // H*W, exact in fp32
    if (lane == 0) {
        out[0] = s * scale;                 // real part
        if (out_size > 1) out[1] = 0.0f;    // imaginary part (exactly zero)
    }
    // Defensively zero any additional output slots.
    for (int i = 2 + lane; i < out_size; i += 32) {
        out[i] = 0.0f;
    }
}

extern "C" void kernel_launch(void* const* d_in, const int* in_sizes, int n_in,
                              void* d_out, int out_size, void* d_ws, size_t ws_size,
                              hipStream_t stream) {
    (void)n_in; (void)d_ws; (void)ws_size;
    const float* x = (const float*)d_in[0];
    float* out = (float*)d_out;

    // Shape is fixed by the reference: (32, 2048, 2048) f32.
    const long long batch_stride = 2048LL * 2048LL;

    fft2_sum_kernel<<<1, 32, 0, stream>>>(x, out, out_size, batch_stride);
    (void)in_sizes;
}